// DiscreteDiagSheafDiffusion_49340584296530
// MI455X (gfx1250) — compile-verified
//
#include <hip/hip_runtime.h>
#include <cstdint>

// Problem constants (from reference)
#define N_NODES 50000
#define IN_CH   128
#define Hc      32          // hidden channels
#define Dd      2           // stalk dim
#define LAYERS  4
#define EU_CNT  400000
#define E_CNT   (2*EU_CNT)  // 800000 directed edges
#define DH      (Dd*Hc)     // 64
#define ND      (N_NODES*Dd)// 100000 rows of [Hc]
#define OUTC    32

typedef __attribute__((ext_vector_type(2))) float v2f;
typedef __attribute__((ext_vector_type(8))) float v8f;

// -------- fp32 WMMA 16x16x4 helper (CDNA5, wave32) --------
__device__ __forceinline__ v8f wmma4(v2f a, v2f b, v8f c) {
  // (neg_a, A, neg_b, B, c_mod, C, reuse_a, reuse_b)
  return __builtin_amdgcn_wmma_f32_16x16x4_f32(false, a, false, b, (short)0, c,
                                               false, false);
}

__device__ __forceinline__ void atomicAddF(float* p, float v) {
  unsafeAtomicAdd(p, v);  // native global_atomic_add_f32 on AMD
}

// C[M,Nn] = mix(A)[M,K] @ W[Nn,K]^T (+bias) (+ELU), optional dual store (C2).
// If Wl != nullptr, A is viewed as [M/2, 2, K] and the 2x2 stalk mix
//   mixA[2n+d', k] = Wl[d',0]*A[2n,k] + Wl[d',1]*A[2n+1,k]
// is applied while loading the A fragment (fuses the sheaf left-weight).
// One wave per 16x16 tile of C; K stepped by 4 with V_WMMA_F32_16X16X4_F32.
// A-frag: lane holds M=lane&15, VGPR pair = K = 2*(lane>>4)+{0,1}
// B-frag: lane holds N=lane&15, VGPR pair = K = 2*(lane>>4)+{0,1}
// C-frag: VGPR r -> row r + 8*(lane>>4), col = lane&15
__global__ void wmma_gemm_xwT(const float* __restrict__ A,
                              const float* __restrict__ W,
                              const float* __restrict__ bias,
                              const float* __restrict__ Wl,
                              float* __restrict__ C,
                              float* __restrict__ C2,
                              int M, int K, int Nn, int do_elu)
{
  const int lane   = threadIdx.x & 31;
  const int wid    = blockIdx.x * (blockDim.x >> 5) + (threadIdx.x >> 5);
  const int ntiles = Nn >> 4;
  const int mtiles = M  >> 4;
  if (wid >= mtiles * ntiles) return;          // wave-uniform exit: EXEC stays full
  const int mt = wid / ntiles;
  const int nt = wid - mt * ntiles;

  const int arow = (mt << 4) + (lane & 15);
  const int bcol = (nt << 4) + (lane & 15);
  const int kb   = (lane >> 4) << 1;

  const float* Wp = W + (size_t)bcol * K + kb;

  v8f acc = {};
  if (Wl) {
    // fused 2x2 stalk mix on A-load
    const int n  = arow >> 1;
    const int dp = arow & 1;
    const float w0 = Wl[dp * 2 + 0];
    const float w1 = Wl[dp * 2 + 1];
    const float* A0 = A + (size_t)(n * 2) * K + kb;        // row 2n
    const float* A1 = A0 + K;                              // row 2n+1
    for (int k0 = 0; k0 < K; k0 += 4) {
      v2f a0 = *(const v2f*)(A0 + k0);
      v2f a1 = *(const v2f*)(A1 + k0);
      v2f a;
      a.x = w0 * a0.x + w1 * a1.x;
      a.y = w0 * a0.y + w1 * a1.y;
      v2f b = *(const v2f*)(Wp + k0);
      acc = wmma4(a, b, acc);
    }
  } else {
    const float* Ap = A + (size_t)arow * K + kb;
    for (int k0 = 0; k0 < K; k0 += 4) {
      v2f a = *(const v2f*)(Ap + k0);
      v2f b = *(const v2f*)(Wp + k0);
      acc = wmma4(a, b, acc);
    }
  }

  const float bv    = bias ? bias[bcol] : 0.0f;
  const int   rbase = (mt << 4) + ((lane >> 4) << 3);
  #pragma unroll
  for (int r = 0; r < 8; ++r) {
    float v = acc[r] + bv;
    if (do_elu) v = (v > 0.0f) ? v : expm1f(v);
    size_t idx = (size_t)(rbase + r) * Nn + bcol;
    C[idx] = v;
    if (C2) C2[idx] = v;
  }
}

// p[n,k] = Ws[k, :64] . xn[n], q[n,k] = Ws[k, 64:] . xn[n]; also zero Dg
__global__ void sheaf_pq(const float* __restrict__ xcur,  // [N, DH]
                         const float* __restrict__ Ws,    // [2, 128] this layer
                         float* __restrict__ p, float* __restrict__ q,
                         float* __restrict__ Dg)
{
  int n = blockIdx.x * blockDim.x + threadIdx.x;
  if (n >= N_NODES) return;
  const float* xrow = xcur + (size_t)n * DH;
  float p0 = 0.f, p1 = 0.f, q0 = 0.f, q1 = 0.f;
  #pragma unroll 4
  for (int j = 0; j < DH; ++j) {
    float xv = xrow[j];
    p0 += Ws[j]             * xv;
    p1 += Ws[2*DH + j]      * xv;
    q0 += Ws[DH + j]        * xv;
    q1 += Ws[2*DH + DH + j] * xv;
  }
  p[n*2+0] = p0; p[n*2+1] = p1;
  q[n*2+0] = q0; q[n*2+1] = q1;
  Dg[n*2+0] = 0.f; Dg[n*2+1] = 0.f;
}

// F[e,k] = tanh(p[src,k] + q[dst,k]); Dg[src,k] += F^2 (segment_sum over src)
__global__ void edge_F(const int* __restrict__ src, const int* __restrict__ dst,
                       const float* __restrict__ p, const float* __restrict__ q,
                       float* __restrict__ F, float* __restrict__ Dg)
{
  int e = blockIdx.x * blockDim.x + threadIdx.x;
  if (e >= E_CNT) return;
  int s = src[e], t = dst[e];
  float f0 = tanhf(p[s*2+0] + q[t*2+0]);
  float f1 = tanhf(p[s*2+1] + q[t*2+1]);
  F[e*2+0] = f0; F[e*2+1] = f1;
  atomicAddF(&Dg[s*2+0], f0*f0);
  atomicAddF(&Dg[s*2+1], f1*f1);
}

__global__ void node_dinv(const float* __restrict__ Dg,
                          float* __restrict__ dinv, float* __restrict__ diagw)
{
  int i = blockIdx.x * blockDim.x + threadIdx.x;
  if (i >= ND) return;
  float g  = Dg[i];
  float dv = rsqrtf(g + 1.0f);
  dinv[i]  = dv;
  diagw[i] = g * dv * dv;
}

__global__ void y_init(const float* __restrict__ xr,
                       const float* __restrict__ diagw, float* __restrict__ y)
{
  int i = blockIdx.x * blockDim.x + threadIdx.x;
  if (i >= ND * Hc) return;
  y[i] = diagw[i >> 5] * xr[i];
}

// y[src*2+d, c] += -F[e,d]*F[rev,d]*dinv[src,d]*dinv[dst,d] * xr[dst*2+d, c]
// 64 threads per edge (d x c); rev(e) = e +/- EU
__global__ void edge_scatter(const int* __restrict__ src, const int* __restrict__ dst,
                             const float* __restrict__ F, const float* __restrict__ dinv,
                             const float* __restrict__ xr, float* __restrict__ y)
{
  unsigned t = blockIdx.x * blockDim.x + threadIdx.x;
  if (t >= (unsigned)E_CNT * DH) return;
  int j = t & (DH - 1);
  int e = t >> 6;
  int d = j >> 5;
  int c = j & 31;
  int s  = src[e], tt = dst[e];
  int er = (e >= EU_CNT) ? (e - EU_CNT) : (e + EU_CNT);
  float offw = -F[e*2+d] * F[er*2+d] * dinv[s*2+d] * dinv[tt*2+d];
  float val  = offw * xr[(size_t)(tt*2+d) * Hc + c];
  atomicAddF(&y[(size_t)(s*2+d) * Hc + c], val);
}

// x0 = (1+tanh(eps[d])) * x0 - elu(y); x = x0
__global__ void elu_residual(const float* __restrict__ y,
                             const float* __restrict__ eps2,  // eps + l*2
                             float* __restrict__ x0, float* __restrict__ xcur)
{
  int i = blockIdx.x * blockDim.x + threadIdx.x;
  if (i >= ND * Hc) return;
  int dp  = (i >> 5) & 1;
  float v = y[i];
  float e = (v > 0.0f) ? v : expm1f(v);
  float coeff = 1.0f + tanhf(eps2[dp]);
  float nx = coeff * x0[i] - e;
  x0[i]   = nx;
  xcur[i] = nx;
}

extern "C" void kernel_launch(void* const* d_in, const int* in_sizes, int n_in,
                              void* d_out, int out_size, void* d_ws, size_t ws_size,
                              hipStream_t stream) {
  const float* x_in   = (const float*)d_in[0];   // [N, 128]
  const int*   ei     = (const int*)  d_in[1];   // [2, E] (int32 under jax default)
  const float* W1     = (const float*)d_in[2];   // [64, 128]
  const float* b1     = (const float*)d_in[3];   // [64]
  const float* Wsheaf = (const float*)d_in[4];   // [4, 2, 128]
  const float* Wleft  = (const float*)d_in[5];   // [4, 2, 2]
  const float* Wright = (const float*)d_in[6];   // [4, 32, 32]
  const float* eps    = (const float*)d_in[7];   // [4, 2, 1]
  const float* W2     = (const float*)d_in[8];   // [32, 64]
  const float* b2     = (const float*)d_in[9];   // [32]
  float* out = (float*)d_out;                    // [N, 32]

  // ---- workspace carve (floats) ----
  float* ws = (float*)d_ws;
  size_t o = 0;
  float* x_cur = ws + o; o += (size_t)ND * Hc;   // 3.2M
  float* x0    = ws + o; o += (size_t)ND * Hc;   // 3.2M
  float* y     = ws + o; o += (size_t)ND * Hc;   // 3.2M
  float* xr    = ws + o; o += (size_t)ND * Hc;   // 3.2M
  float* p     = ws + o; o += 2 * (size_t)N_NODES;
  float* q     = ws + o; o += 2 * (size_t)N_NODES;
  float* F     = ws + o; o += 2 * (size_t)E_CNT;
  float* Dg    = ws + o; o += (size_t)ND;
  float* dinv  = ws + o; o += (size_t)ND;
  float* diagw = ws + o; o += (size_t)ND;

  const int* src = ei;
  const int* dst = ei + E_CNT;

  const int TPB = 256;                 // 8 wave32s per block
  const int elemBlocks = (ND * Hc + TPB - 1) / TPB;

  // x = elu(x @ W1^T + b1), dual-store into x_cur and x0
  {
    int tiles = (N_NODES / 16) * (DH / 16);
    wmma_gemm_xwT<<<(tiles + 7) / 8, TPB, 0, stream>>>(
        x_in, W1, b1, nullptr, x_cur, x0, N_NODES, IN_CH, DH, 1);
  }

  for (int l = 0; l < LAYERS; ++l) {
    sheaf_pq<<<(N_NODES + TPB - 1) / TPB, TPB, 0, stream>>>(
        x_cur, Wsheaf + (size_t)l * 2 * 2 * DH, p, q, Dg);
    edge_F<<<(E_CNT + TPB - 1) / TPB, TPB, 0, stream>>>(src, dst, p, q, F, Dg);
    node_dinv<<<(ND + TPB - 1) / TPB, TPB, 0, stream>>>(Dg, dinv, diagw);
    // xr = (stalk-mixed x) @ W_right^T  (left-weight fused into A fragment load)
    {
      int tiles = (ND / 16) * (Hc / 16);
      wmma_gemm_xwT<<<(tiles + 7) / 8, TPB, 0, stream>>>(
          x_cur, Wright + (size_t)l * Hc * Hc, nullptr, Wleft + (size_t)l * 4,
          xr, nullptr, ND, Hc, Hc, 0);
    }
    y_init<<<elemBlocks, TPB, 0, stream>>>(xr, diagw, y);
    edge_scatter<<<((unsigned)E_CNT * DH + TPB - 1) / TPB, TPB, 0, stream>>>(
        src, dst, F, dinv, xr, y);
    elu_residual<<<elemBlocks, TPB, 0, stream>>>(y, eps + (size_t)l * 2, x0, x_cur);
  }

  // out = x @ W2^T + b2
  {
    int tiles = (N_NODES / 16) * (OUTC / 16);
    wmma_gemm_xwT<<<(tiles + 7) / 8, TPB, 0, stream>>>(
        x_cur, W2, b2, nullptr, out, nullptr, N_NODES, DH, OUTC, 0);
  }
}